// newGCN_429496729849
// MI455X (gfx1250) — compile-verified
//
#include <hip/hip_runtime.h>

#define N_NODES 50000
#define N_EDGES 800000
#define HDIM 128
#define NGRAPH 64
#define NOUT 10
#define NLAYER 6
#define BN_EPS 1e-5f

typedef float v2f __attribute__((ext_vector_type(2)));
typedef float v8f __attribute__((ext_vector_type(8)));

// LDS geometry (64 banks x 4B):
//  A tile: 128 rows x 32 K, stride 34 -> A-frag b64 loads hit banks
//    2*(17r mod 32)+k : distinct per lane, conflict-free.
//  B tile: row-pair interleaved, Bs[(k>>1)*BPAIR + 2c + (k&1)], BPAIR=288.
//    B-frag is then a single b64 at p*288 + 2c: lanes 0-15 cover banks
//    {b0+2i, +1}; lanes 16-31 use pair p+1 -> +288 -> bank shift 288%64=32,
//    the opposite half of the banks. Conflict-free b64 loads.
#define ASTR  34
#define BPAIR 288

__global__ __launch_bounds__(256) void zero_kernel(float* p, int n) {
  int i = blockIdx.x * 256 + threadIdx.x;
  if (i < n) p[i] = 0.0f;
}

__global__ __launch_bounds__(256) void initdeg_kernel(float* deg, int n) {
  int i = blockIdx.x * 256 + threadIdx.x;
  if (i < n) deg[i] = 1.0f;   // self-loop contributes 1
}

__global__ __launch_bounds__(256) void countdeg_kernel(const int* __restrict__ dst,
                                                       float* deg, int e) {
  int i = blockIdx.x * 256 + threadIdx.x;
  if (i < e) unsafeAtomicAdd(&deg[dst[i]], 1.0f);
}

__global__ __launch_bounds__(256) void dinv_kernel(float* deg_selfw, float* dinv, int n) {
  int i = blockIdx.x * 256 + threadIdx.x;
  if (i < n) {
    float d = deg_selfw[i];
    dinv[i] = rsqrtf(d);
    deg_selfw[i] = 1.0f / d;  // self-loop weight dinv^2, overwrites deg in place
  }
}

// C[M x 128] = act(A)[M x 128] @ W[128 x 128] (+bias)(+relu out), fp32 WMMA.
// act(A) = relu(A*aScale[c] + aShift[c]) if aScale != nullptr (fused BN+ReLU
// of the previous GCN layer), else identity.
// 256 threads = 8 wave32; wave w computes rows [blk*128 + 16w, +16), all cols.
__global__ __launch_bounds__(256)
void gemm128_wmma(const float* __restrict__ A, const float* __restrict__ W,
                  const float* __restrict__ bias, float* __restrict__ C,
                  int M, int doRelu,
                  const float* __restrict__ aScale, const float* __restrict__ aShift) {
  __shared__ float As[128 * ASTR];  // 17,408 B
  __shared__ float Bs[16 * BPAIR];  // 18,432 B (16 row-pairs)

  const int tid  = threadIdx.x;
  const int wave = tid >> 5;
  const int lane = tid & 31;
  const int rowBase = blockIdx.x * 128;
  const int colLane = lane & 15;
  const int khalf   = (lane >> 4) << 1;        // 0 or 2 (K-pair select)
  const int mrow    = (wave << 4) + colLane;   // A-matrix row for this lane

  v8f acc[8] = {};

  for (int kb = 0; kb < 128; kb += 32) {
    __syncthreads();
    // A tile 128x32 as float2, optional fused BN+ReLU on the fly
    #pragma unroll
    for (int it = 0; it < 8; ++it) {
      int idx = tid + (it << 8);
      int r = idx >> 4, c2 = (idx & 15) << 1;
      float2 v = make_float2(0.0f, 0.0f);
      int gr = rowBase + r;
      if (gr < M) v = *(const float2*)(A + gr * HDIM + kb + c2);
      if (aScale) {
        int c = kb + c2;
        v.x = fmaxf(fmaf(v.x, aScale[c],     aShift[c]),     0.0f);
        v.y = fmaxf(fmaf(v.y, aScale[c + 1], aShift[c + 1]), 0.0f);
      }
      *(float2*)(&As[r * ASTR + c2]) = v;
    }
    // W tile 32x128, stored row-pair interleaved for b64 B-fragments
    #pragma unroll
    for (int it = 0; it < 4; ++it) {
      int idx = tid + (it << 8);
      int r = idx >> 5;               // 0..31 (uniform per wave)
      int c = (idx & 31) << 2;        // 0..124
      float4 v = *(const float4*)(W + (kb + r) * HDIM + c);
      float* bp = &Bs[(r >> 1) * BPAIR + (c << 1) + (r & 1)];
      bp[0] = v.x; bp[2] = v.y; bp[4] = v.z; bp[6] = v.w;
    }
    __syncthreads();

    for (int kk = 0; kk < 32; kk += 4) {
      // A 16x4 frag: VGPR0 = K={0|2}, VGPR1 = K={1|3} per lane half
      v2f a = *(const v2f*)(&As[mrow * ASTR + kk + khalf]);
      const int p = (kk + khalf) >> 1;  // even row -> pair index
      #pragma unroll
      for (int nt = 0; nt < 8; ++nt) {
        // B 4x16 frag: rows k,k+1 at col -> adjacent in interleaved layout
        v2f b = *(const v2f*)(&Bs[p * BPAIR + (((nt << 4) + colLane) << 1)]);
        acc[nt] = __builtin_amdgcn_wmma_f32_16x16x4_f32(
            false, a, false, b, (short)0, acc[nt], false, false);
      }
    }
  }

  // C/D layout: VGPR v -> M = v (lanes 0-15) or v+8 (lanes 16-31), N = lane&15
  const int rowOff = (lane >> 4) << 3;
  #pragma unroll
  for (int nt = 0; nt < 8; ++nt) {
    int col = (nt << 4) + colLane;
    float bv = bias ? bias[col] : 0.0f;
    #pragma unroll
    for (int v = 0; v < 8; ++v) {
      int gr = rowBase + (wave << 4) + rowOff + v;
      if (gr < M) {
        float r = acc[nt][v] + bv;
        if (doRelu) r = fmaxf(r, 0.0f);
        C[gr * HDIM + col] = r;
      }
    }
  }
}

// One wave per edge: 32 lanes x float4 gather of t[src], atomic add into agg[dst].
__global__ __launch_bounds__(256)
void scatter_kernel(const float* __restrict__ t, const int* __restrict__ src,
                    const int* __restrict__ dst, const float* __restrict__ dinv,
                    float* agg, int e) {
  int gid  = blockIdx.x * 256 + threadIdx.x;
  int eidx = gid >> 5;
  if (eidx >= e) return;
  int lane = gid & 31;
  int s = src[eidx], d = dst[eidx];
  float w = dinv[s] * dinv[d];
  const float4 v = *(const float4*)(t + s * HDIM + (lane << 2));
  float* o = agg + d * HDIM + (lane << 2);
  unsafeAtomicAdd(o + 0, v.x * w);
  unsafeAtomicAdd(o + 1, v.y * w);
  unsafeAtomicAdd(o + 2, v.z * w);
  unsafeAtomicAdd(o + 3, v.w * w);
}

// h = agg + t*selfw + bias (pre-BN); accumulate per-channel sum/sumsq with one
// atomic per channel per 64-row block (~200k atomics instead of 13M).
__global__ __launch_bounds__(128)
void combine_kernel(const float* __restrict__ agg, const float* __restrict__ t,
                    const float* __restrict__ selfw, const float* __restrict__ bias,
                    float* __restrict__ h, float* sum, float* sumsq, int n, int doStats) {
  int c  = threadIdx.x;
  int r0 = blockIdx.x << 6;
  float b = bias[c];
  float s = 0.0f, s2 = 0.0f;
  for (int i = 0; i < 64; ++i) {
    int r = r0 + i;
    if (r >= n) break;
    float v = agg[r * HDIM + c] + t[r * HDIM + c] * selfw[r] + b;
    h[r * HDIM + c] = v;
    s += v; s2 += v * v;
  }
  if (doStats) {
    unsafeAtomicAdd(&sum[c], s);
    unsafeAtomicAdd(&sumsq[c], s2);
  }
}

__global__ __launch_bounds__(128)
void bnfinal_kernel(const float* sum, const float* sumsq, const float* gamma,
                    const float* beta, float* scale, float* shift, float invn) {
  int c = threadIdx.x;
  float mu  = sum[c] * invn;
  float var = sumsq[c] * invn - mu * mu;   // biased variance
  float sc  = rsqrtf(var + BN_EPS) * gamma[c];
  scale[c] = sc;
  shift[c] = beta[c] - mu * sc;
}

__global__ __launch_bounds__(256)
void poolcount_kernel(const int* batch, float* counts, int n) {
  int i = blockIdx.x * 256 + threadIdx.x;
  if (i < n) unsafeAtomicAdd(&counts[batch[i]], 1.0f);
}

__global__ __launch_bounds__(256)
void poolsum_kernel(const float* __restrict__ h, const int* __restrict__ batch,
                    float* pooled, int n) {
  int i = blockIdx.x * 256 + threadIdx.x;
  if (i < n) {
    int node = i >> 7, c = i & (HDIM - 1);
    unsafeAtomicAdd(&pooled[batch[node] * HDIM + c], h[i]);
  }
}

__global__ __launch_bounds__(128)
void pooldiv_kernel(float* pooled, const float* counts) {
  int g = blockIdx.x, c = threadIdx.x;
  pooled[g * HDIM + c] /= fmaxf(counts[g], 1.0f);
}

__global__ __launch_bounds__(128)
void headbn_kernel(float* z, const float* gamma, const float* beta) {
  int c = threadIdx.x;
  float s = 0.0f, s2 = 0.0f;
  for (int g = 0; g < NGRAPH; ++g) { float v = z[g * HDIM + c]; s += v; s2 += v * v; }
  float mu  = s * (1.0f / NGRAPH);
  float var = s2 * (1.0f / NGRAPH) - mu * mu;
  float sc  = rsqrtf(var + BN_EPS) * gamma[c];
  float sh  = beta[c] - mu * sc;
  for (int g = 0; g < NGRAPH; ++g)
    z[g * HDIM + c] = fmaxf(z[g * HDIM + c] * sc + sh, 0.0f);
}

__global__ __launch_bounds__(128)
void final_kernel(const float* __restrict__ z, const float* __restrict__ W3,
                  const float* __restrict__ b3, float* out) {
  int i = blockIdx.x * 128 + threadIdx.x;
  if (i >= NGRAPH * NOUT) return;
  int g = i / NOUT, o = i % NOUT;
  float acc = b3[o];
  for (int k = 0; k < HDIM; ++k) acc += z[g * HDIM + k] * W3[k * NOUT + o];
  out[i] = acc;
}

extern "C" void kernel_launch(void* const* d_in, const int* in_sizes, int n_in,
                              void* d_out, int out_size, void* d_ws, size_t ws_size,
                              hipStream_t stream) {
  const float* x     = (const float*)d_in[0];
  const int*   ei    = (const int*)d_in[1];
  const int*   batch = (const int*)d_in[2];
  const float* convW = (const float*)d_in[3];
  const float* convB = (const float*)d_in[4];
  const float* bnG   = (const float*)d_in[5];
  const float* bnB   = (const float*)d_in[6];
  const float* hW1   = (const float*)d_in[7];
  const float* hb1   = (const float*)d_in[8];
  const float* hbnG  = (const float*)d_in[9];
  const float* hbnB  = (const float*)d_in[10];
  const float* hW2   = (const float*)d_in[11];
  const float* hb2   = (const float*)d_in[12];
  const float* hW3   = (const float*)d_in[13];
  const float* hb3   = (const float*)d_in[14];

  const int N = N_NODES, E = N_EDGES, H = HDIM, G = NGRAPH;
  const int* srcIdx = ei;
  const int* dstIdx = ei + E;

  // Workspace layout (~78 MB of fp32)
  float* ws    = (float*)d_ws;
  float* selfw = ws;  ws += N;             // deg, then 1/deg
  float* dinv  = ws;  ws += N;             // deg^{-1/2}
  float* hbuf  = ws;  ws += (size_t)N * H; // node features (pre-BN per layer)
  float* tbuf  = ws;  ws += (size_t)N * H; // act(h) @ W
  float* agg   = ws;  ws += (size_t)N * H; // scatter accumulator
  float* sum   = ws;  ws += H;
  float* sumsq = ws;  ws += H;             // contiguous with sum (zeroed together)
  float* scale = ws;  ws += H;
  float* shift = ws;  ws += H;
  float* pooled= ws;  ws += G * H;
  float* counts= ws;  ws += G;             // contiguous with pooled (zeroed together)
  float* z1    = ws;  ws += G * H;
  float* z2    = ws;  ws += G * H;

  auto cdiv = [](int a, int b) { return (a + b - 1) / b; };

  // degree -> dinv, selfw
  initdeg_kernel<<<cdiv(N, 256), 256, 0, stream>>>(selfw, N);
  countdeg_kernel<<<cdiv(E, 256), 256, 0, stream>>>(dstIdx, selfw, E);
  dinv_kernel<<<cdiv(N, 256), 256, 0, stream>>>(selfw, dinv, N);

  const float* hin = x;
  for (int l = 0; l < NLAYER; ++l) {
    // GEMM consumes previous layer's BN+ReLU on the fly (l>0); layer 0 raw x.
    const float* sc = (l > 0) ? scale : nullptr;
    const float* sh = (l > 0) ? shift : nullptr;
    gemm128_wmma<<<cdiv(N, 128), 256, 0, stream>>>(
        hin, convW + (size_t)l * H * H, nullptr, tbuf, N, 0, sc, sh);
    zero_kernel<<<cdiv(N * H, 256), 256, 0, stream>>>(agg, N * H);
    scatter_kernel<<<cdiv(E * 32, 256), 256, 0, stream>>>(
        tbuf, srcIdx, dstIdx, dinv, agg, E);
    int doStats = (l < NLAYER - 1);
    if (doStats) zero_kernel<<<1, 256, 0, stream>>>(sum, 2 * H);
    combine_kernel<<<cdiv(N, 64), 128, 0, stream>>>(
        agg, tbuf, selfw, convB + l * H, hbuf, sum, sumsq, N, doStats);
    if (doStats) {
      bnfinal_kernel<<<1, 128, 0, stream>>>(
          sum, sumsq, bnG + l * H, bnB + l * H, scale, shift, 1.0f / N);
    }
    hin = hbuf;
  }

  // global mean pool (layer 5 has no BN -> hbuf is final activation)
  zero_kernel<<<cdiv(G * H + G, 256), 256, 0, stream>>>(pooled, G * H + G);
  poolcount_kernel<<<cdiv(N, 256), 256, 0, stream>>>(batch, counts, N);
  poolsum_kernel<<<cdiv(N * H, 256), 256, 0, stream>>>(hbuf, batch, pooled, N * H);
  pooldiv_kernel<<<G, 128, 0, stream>>>(pooled, counts);

  // head: Linear -> BN+ReLU -> Linear+ReLU -> Linear
  gemm128_wmma<<<1, 256, 0, stream>>>(pooled, hW1, hb1, z1, G, 0, nullptr, nullptr);
  headbn_kernel<<<1, 128, 0, stream>>>(z1, hbnG, hbnB);
  gemm128_wmma<<<1, 256, 0, stream>>>(z1, hW2, hb2, z2, G, 1, nullptr, nullptr);
  final_kernel<<<cdiv(G * NOUT, 128), 128, 0, stream>>>(z2, hW3, hb3, (float*)d_out);
}